// HierKVBank_42545946034838
// MI455X (gfx1250) — compile-verified
//
#include <hip/hip_runtime.h>
#include <hip/hip_bf16.h>
#include <math.h>

typedef __attribute__((ext_vector_type(2))) float v2f;
typedef __attribute__((ext_vector_type(8))) float v8f;

#define B_DIM 4
#define T_DIM 8192
#define C_DIM 1024
#define W0_N 128
#define W1_N 64
#define W2_N 16

// ---------------------------------------------------------------------------
// Generic stride-2 pooling kernel. One wave (32 lanes) handles 16 input rows
// (= 8 chunks) of one batch b.
// Phase 1: per-row dot(q, row) via V_WMMA_F32_16X16X4_F32.
//   A (16x4 f32): lanes 0-15 hold rows M=0..15 cols {c,c+1}; lanes 16-31 cols
//   {c+2,c+3}  (documented A layout). B (4x16): q[c..c+3] broadcast across all
//   16 columns, so every accumulator column holds the same 16 row-dots.
// Phase 2: softmax weights, V-row squared-difference reduction for recon,
//   and pooled sk/sv writes for the tail window.
// ---------------------------------------------------------------------------
__global__ void hierkv_pool_kernel(
    const float* __restrict__ Kin, const float* __restrict__ Vin,
    const float* __restrict__ q,
    int rowsPerB,        // rows per batch (level1: 8064, level2: 64)
    long bStrideIn,      // element stride between batches in Kin/Vin
    int nChunks,         // chunks per batch = rowsPerB/2
    int tailCount,       // number of tail chunks written to Kout/Vout
    float* __restrict__ Kout, float* __restrict__ Vout,  // (B, tailCount, C)
    float reconScale,
    float* __restrict__ reconAcc)
{
    const int lane = threadIdx.x & 31;
    const int waveGlobal = (int)((blockIdx.x * blockDim.x + threadIdx.x) >> 5);
    const int wavesPerB = rowsPerB >> 4;
    const int b = waveGlobal / wavesPerB;
    if (b >= B_DIM) return;                       // wave-uniform guard
    const int rowBase = (waveGlobal % wavesPerB) << 4;

    const float* Kb = Kin + (long)b * bStrideIn;
    const float* Vb = Vin + (long)b * bStrideIn;

    // ---------------- phase 1: scores via WMMA f32 16x16x4 ----------------
    const int r = rowBase + (lane & 15);
    const int colSel = (lane >> 4) << 1;          // 0 for lanes 0-15, 2 for 16-31
    const float* Arow = Kb + (long)r * C_DIM + colSel;
    const float* qp   = q + colSel;

    v8f acc = {};
    for (int c = 0; c < C_DIM; c += 4) {
        v2f a, bq;
        a.x  = Arow[c];
        a.y  = Arow[c + 1];
        bq.x = qp[c];
        bq.y = qp[c + 1];
        acc = __builtin_amdgcn_wmma_f32_16x16x4_f32(
            /*neg_a=*/false, a, /*neg_b=*/false, bq,
            /*c_mod=*/(short)0, acc, /*reuse_a=*/false, /*reuse_b=*/false);
    }

    // D layout: VGPR m, lanes 0-15 -> M=m (rows rowBase+0..7),
    //           lanes 16-31 -> M=m+8 (rows rowBase+8..15); all N identical.
    float score[16];
#pragma unroll
    for (int i = 0; i < 8; ++i) {
        score[i]     = __shfl(acc[i], 0, 32);
        score[i + 8] = __shfl(acc[i], 16, 32);
    }

    // softmax over each stride-2 pair
    float w0a[8], w1a[8];
#pragma unroll
    for (int j = 0; j < 8; ++j) {
        float s0 = score[2 * j], s1 = score[2 * j + 1];
        float m  = fmaxf(s0, s1);
        float e0 = __expf(s0 - m);
        float e1 = __expf(s1 - m);
        float inv = 1.0f / (e0 + e1);
        w0a[j] = e0 * inv;
        w1a[j] = e1 * inv;
    }

    // ---------------- phase 2: recon + tail pooled outputs ----------------
    float reconLocal = 0.0f;
    for (int j = 0; j < 8; ++j) {
        const int chunk = (rowBase >> 1) + j;
        const int t0 = rowBase + 2 * j;
        const float w0 = w0a[j], w1 = w1a[j];

        const float* v0p = Vb + (long)t0 * C_DIM;
        const float* v1p = v0p + C_DIM;
        const float* k0p = Kb + (long)t0 * C_DIM;
        const float* k1p = k0p + C_DIM;

        const bool tail = (chunk >= nChunks - tailCount);
        const int  p    = chunk - (nChunks - tailCount);
        float* ko = Kout + ((long)b * tailCount + p) * C_DIM;
        float* vo = Vout + ((long)b * tailCount + p) * C_DIM;

        float ssd = 0.0f;
        for (int c0 = lane * 4; c0 < C_DIM; c0 += 128) {
            float4 v0 = *(const float4*)(v0p + c0);
            float4 v1 = *(const float4*)(v1p + c0);
            float dx = v0.x - v1.x, dy = v0.y - v1.y;
            float dz = v0.z - v1.z, dw = v0.w - v1.w;
            ssd += dx * dx + dy * dy + dz * dz + dw * dw;
            if (tail) {
                float4 k0 = *(const float4*)(k0p + c0);
                float4 k1 = *(const float4*)(k1p + c0);
                float4 kr, vr;
                kr.x = w0 * k0.x + w1 * k1.x;  kr.y = w0 * k0.y + w1 * k1.y;
                kr.z = w0 * k0.z + w1 * k1.z;  kr.w = w0 * k0.w + w1 * k1.w;
                vr.x = w0 * v0.x + w1 * v1.x;  vr.y = w0 * v0.y + w1 * v1.y;
                vr.z = w0 * v0.z + w1 * v1.z;  vr.w = w0 * v0.w + w1 * v1.w;
                *(float4*)(ko + c0) = kr;
                *(float4*)(vo + c0) = vr;
            }
        }
        // wave butterfly reduction of ssd
#pragma unroll
        for (int off = 16; off; off >>= 1) ssd += __shfl_xor(ssd, off, 32);

        const float d = w0 - 0.5f;
        reconLocal += d * d * ssd;
    }

    if (lane == 0) atomicAdd(reconAcc, reconLocal * reconScale);
}

// Copy K0 = K[:, -128:, :] and V0 = V[:, -128:, :] (float4-vectorized).
__global__ void hierkv_copy_tail_kernel(const float* __restrict__ K,
                                        const float* __restrict__ V,
                                        float* __restrict__ out)
{
    const long NPER4 = (long)B_DIM * W0_N * C_DIM / 4;   // 131072 float4 per tensor
    long i4 = (long)blockIdx.x * blockDim.x + threadIdx.x;
    if (i4 >= 2 * NPER4) return;
    const bool isV = (i4 >= NPER4);
    const long j4 = isV ? (i4 - NPER4) : i4;
    const long e  = j4 * 4;                              // flat element in (B,128,C)
    const long perB = (long)W0_N * C_DIM;
    const long b    = e / perB;
    const long rem  = e % perB;                          // t*C + c, t in [0,128)
    const float* src = (isV ? V : K) + b * (long)T_DIM * C_DIM
                     + (long)(T_DIM - W0_N) * C_DIM + rem;
    float* dst = out + (isV ? (long)B_DIM * W0_N * C_DIM : 0) + e;
    *(float4*)dst = *(const float4*)src;
}

__global__ void hierkv_zero_kernel(float* __restrict__ acc) { acc[0] = 0.0f; }

__global__ void hierkv_finish_kernel(const float* __restrict__ acc,
                                     float* __restrict__ dst) { dst[0] = acc[0]; }

extern "C" void kernel_launch(void* const* d_in, const int* in_sizes, int n_in,
                              void* d_out, int out_size, void* d_ws, size_t ws_size,
                              hipStream_t stream) {
    const float* K = (const float*)d_in[0];   // (4, 8192, 1024) f32
    const float* V = (const float*)d_in[1];   // (4, 8192, 1024) f32
    const float* q = (const float*)d_in[2];   // (2, 1024) f32
    float* out = (float*)d_out;
    float* acc = (float*)d_ws;                // 1-float recon accumulator

    // output layout (flat float offsets, return order)
    const long K1_off = 2L * B_DIM * W0_N * C_DIM;            // 1048576
    const long V1_off = K1_off + (long)B_DIM * W1_N * C_DIM;  // 1310720
    const long K2_off = V1_off + (long)B_DIM * W1_N * C_DIM;  // 1572864
    const long V2_off = K2_off + (long)B_DIM * W2_N * C_DIM;  // 1638400
    const long R_off  = V2_off + (long)B_DIM * W2_N * C_DIM;  // 1703936

    hipLaunchKernelGGL(hierkv_zero_kernel, dim3(1), dim3(1), 0, stream, acc);

    // K0 / V0 copies: 262144 float4 -> 1024 blocks x 256
    hipLaunchKernelGGL(hierkv_copy_tail_kernel, dim3(1024), dim3(256), 0, stream,
                       K, V, out);

    // level 1: rowsPerB=8064 -> 504 waves/batch, 2016 waves = 252 blocks x 256
    hipLaunchKernelGGL(hierkv_pool_kernel, dim3(252), dim3(256), 0, stream,
                       K, V, q,
                       8064, (long)T_DIM * C_DIM, 4032, W1_N,
                       out + K1_off, out + V1_off,
                       1.0f / (4032.0f * (float)B_DIM * (float)C_DIM), acc);

    // level 2: operates on K1/V1 just written in d_out; 16 waves = 2 blocks x 256
    hipLaunchKernelGGL(hierkv_pool_kernel, dim3(2), dim3(256), 0, stream,
                       out + K1_off, out + V1_off, q + C_DIM,
                       64, (long)W1_N * C_DIM, 32, W2_N,
                       out + K2_off, out + V2_off,
                       0.5f / ((float)B_DIM * (float)C_DIM), acc);

    hipLaunchKernelGGL(hierkv_finish_kernel, dim3(1), dim3(1), 0, stream,
                       acc, out + R_off);
}